// GIN_83391085019876
// MI455X (gfx1250) — compile-verified
//
#include <hip/hip_runtime.h>
#include <hip/hip_bf16.h>

// ---------------------------------------------------------------------------
// GIN forward for MI455X (gfx1250), fp32 WMMA path.
// Memory-bound workload (~6.6GB min traffic @ 23.3TB/s): strategy is
//  (1) read a1 exactly twice (min + bit-pack mask into 10.6MB, L2-resident),
//  (2) all GEMMs on v_wmma_f32_16x16x4_f32 (fp32 free at this intensity),
//  (3) fuse BatchNorm: GEMM epilogues accumulate column sum/sumsq, consumers
//      apply relu(x*scale+shift) during A-load / LDS fill.
// ---------------------------------------------------------------------------

#define Bb 8
#define Tt 64
#define Nn 400
#define Cin 400
#define Hh 128
#define BT (Bb*Tt)            // 512 blocks
#define MROWS (BT*Nn)         // 204800 total rows
#define WPR 13                // 32-bit mask words per 400-bit row

typedef __attribute__((ext_vector_type(2))) float v2f;
typedef __attribute__((ext_vector_type(8))) float v8f;

__device__ inline v8f wmma4_f32(v2f a, v2f b, v8f c) {
  // D = A(16x4,f32) * B(4x16,f32) + C(16x16,f32)
  return __builtin_amdgcn_wmma_f32_16x16x4_f32(
      /*neg_a=*/false, a, /*neg_b=*/false, b,
      /*c_mod=*/(short)0, c, /*reuse_a=*/false, /*reuse_b=*/false);
}

__device__ inline v8f v8f_zero() {
  v8f z = {0.f,0.f,0.f,0.f,0.f,0.f,0.f,0.f};
  return z;
}

// ---------------------------------------------------------------------------
// Per-(b,t) min over the 400x400 adjacency block.
__global__ void k_blockmin(const float* __restrict__ a, float* __restrict__ bmin) {
  __shared__ float red[256];
  const int bt = blockIdx.x;
  const float* p = a + (size_t)bt * (Nn*Nn);
  float m = 3.402823466e38f;
  for (int i = threadIdx.x; i < Nn*Nn; i += 256) m = fminf(m, p[i]);
  red[threadIdx.x] = m;
  __syncthreads();
  for (int s = 128; s > 0; s >>= 1) {
    if (threadIdx.x < s) red[threadIdx.x] = fminf(red[threadIdx.x], red[threadIdx.x+s]);
    __syncthreads();
  }
  if (threadIdx.x == 0) bmin[bt] = red[0];
}

// ---------------------------------------------------------------------------
// Pack mask bits: bit j of word (bt,row,wi) = (a1[bt,row,wi*32+j] > blockmin).
__global__ void k_maskpack(const float* __restrict__ a, const float* __restrict__ bmin,
                           unsigned* __restrict__ packed) {
  const int bt = blockIdx.x;
  const float mn = bmin[bt];
  const float* p = a + (size_t)bt*Nn*Nn;
  unsigned* mp = packed + (size_t)bt*Nn*WPR;
  for (int w = threadIdx.x; w < Nn*WPR; w += 256) {
    const int row = w / WPR, wi = w - row*WPR;
    const int j0 = wi*32;
    const int jmax = (Nn - j0) < 32 ? (Nn - j0) : 32;
    const float* pr = p + (size_t)row*Nn + j0;
    unsigned bits = 0;
    for (int b = 0; b < jmax; ++b) if (pr[b] > mn) bits |= (1u << b);
    mp[w] = bits;
  }
}

// ---------------------------------------------------------------------------
// h0 = v1 @ W0 + b0.  M=204800, K=400, N=128.  K-chunk 80 -> W tile in LDS.
__global__ void k_gemm0(const float* __restrict__ A, const float* __restrict__ W,
                        const float* __restrict__ bias, float* __restrict__ out) {
  __shared__ float Wlds[80*129];
  const int tid  = threadIdx.x;
  const int wave = tid >> 5, lane = tid & 31;
  const int lrow = lane & 15;
  const int off2 = (lane >> 4) << 1;           // 0 or 2
  const int rowbase = blockIdx.x * 128 + wave * 16;
  const int myrow = rowbase + lrow;

  v8f acc[8];
#pragma unroll
  for (int t = 0; t < 8; ++t) acc[t] = v8f_zero();

  for (int kc = 0; kc < 5; ++kc) {
    const int kbase = kc * 80;
    __syncthreads();
    for (int i = tid; i < 80*128; i += 256) {
      const int r = i >> 7, c = i & 127;
      Wlds[r*129 + c] = W[(size_t)(kbase + r)*Hh + c];
    }
    __syncthreads();

    v2f afr[20];
    const float* arow = A + (size_t)myrow*Cin + kbase + off2;
#pragma unroll
    for (int ks = 0; ks < 20; ++ks) {
      afr[ks].x = arow[ks*4 + 0];
      afr[ks].y = arow[ks*4 + 1];
    }
#pragma unroll
    for (int tn = 0; tn < 8; ++tn) {
      const int col = tn*16 + lrow;
#pragma unroll
      for (int ks = 0; ks < 20; ++ks) {
        v2f bfr;
        bfr.x = Wlds[(ks*4 + off2    )*129 + col];
        bfr.y = Wlds[(ks*4 + off2 + 1)*129 + col];
        acc[tn] = wmma4_f32(afr[ks], bfr, acc[tn]);
      }
    }
  }
  const int mofs = (lane >> 4) * 8;
#pragma unroll
  for (int tn = 0; tn < 8; ++tn) {
    const int col = tn*16 + lrow;
    const float bv = bias[col];
#pragma unroll
    for (int i = 0; i < 8; ++i)
      out[(size_t)(rowbase + mofs + i)*Hh + col] = acc[tn][i] + bv;
  }
}

// ---------------------------------------------------------------------------
// Block-diagonal aggregation: out = mask(bt) @ h(bt) + eps*h(bt).
// A synthesized from packed bits (L2), B tile (400x16 feature slice) in LDS,
// with optional fused BN+ReLU transform applied while filling LDS.
__global__ void k_agg(const unsigned* __restrict__ mask, const float* __restrict__ hin,
                      float* __restrict__ out, const float* __restrict__ eps_p, int layer,
                      const float* __restrict__ tsc, const float* __restrict__ tsh,
                      int do_transform) {
  __shared__ float Hl[Nn*16];                   // 25.6 KB feature slice
  const int bt = blockIdx.x;
  const int tileN = blockIdx.y;                 // 0..7
  const int tid = threadIdx.x;
  const float epsv = eps_p[layer];

  // fill LDS feature slice (with fused BN+ReLU on layer 1)
  for (int i = tid; i < Nn*16; i += 256) {
    const int node = i >> 4, c16 = i & 15;
    const int col = tileN*16 + c16;
    float v = hin[((size_t)bt*Nn + node)*Hh + col];
    if (do_transform) v = fmaxf(0.f, fmaf(v, tsc[col], tsh[col]));
    Hl[node*16 + c16] = v;
  }
  __syncthreads();

  const int wave = tid >> 5, lane = tid & 31;
  const int lrow = lane & 15;
  const int off2 = (lane >> 4) << 1;
  const int col  = tileN*16 + lrow;
  const unsigned* mbase = mask + (size_t)bt*Nn*WPR;
  const int mofs = (lane >> 4) * 8;

  for (int tm = wave; tm < 25; tm += 8) {       // 25 M-tiles round-robin
    const int arow = tm*16 + lrow;              // node row feeding A
    const unsigned* mr = mbase + (size_t)arow*WPR;
    v8f acc = v8f_zero();

    for (int wi = 0; wi < 12; ++wi) {           // 8 k-steps per mask word
      const unsigned wb = mr[wi];
      const int kw = wi * 32;
#pragma unroll
      for (int kk = 0; kk < 8; ++kk) {
        const int sh = kk*4 + off2;
        v2f afr;
        afr.x = ((wb >> sh)     & 1u) ? 1.f : 0.f;
        afr.y = ((wb >> (sh+1)) & 1u) ? 1.f : 0.f;
        const int krow = kw + kk*4 + off2;
        v2f bfr;
        bfr.x = Hl[krow*16 + lrow];
        bfr.y = Hl[(krow+1)*16 + lrow];
        acc = wmma4_f32(afr, bfr, acc);
      }
    }
    {                                           // tail: word 12, bits 0..15
      const unsigned wb = mr[12];
#pragma unroll
      for (int kk = 0; kk < 4; ++kk) {
        const int sh = kk*4 + off2;
        v2f afr;
        afr.x = ((wb >> sh)     & 1u) ? 1.f : 0.f;
        afr.y = ((wb >> (sh+1)) & 1u) ? 1.f : 0.f;
        const int krow = 384 + kk*4 + off2;
        v2f bfr;
        bfr.x = Hl[krow*16 + lrow];
        bfr.y = Hl[(krow+1)*16 + lrow];
        acc = wmma4_f32(afr, bfr, acc);
      }
    }
#pragma unroll
    for (int i = 0; i < 8; ++i) {
      const int nrow = tm*16 + mofs + i;
      const float hval = Hl[nrow*16 + lrow];    // transformed h for eps-term
      out[((size_t)bt*Nn + nrow)*Hh + col] = acc[i] + epsv * hval;
    }
  }
}

// ---------------------------------------------------------------------------
// y = f(A) @ W + b, f = identity or fused relu(x*scale+shift) (prev BN).
// Also accumulates per-column sum / sum-of-squares for this layer's BN.
// M=204800, K=128, N=128. K-chunk 64.
__global__ void k_gemm_mlp(const float* __restrict__ A, const float* __restrict__ W,
                           const float* __restrict__ bias, float* __restrict__ out,
                           const float* __restrict__ tsc, const float* __restrict__ tsh,
                           int do_transform,
                           float* __restrict__ gsum, float* __restrict__ gsq) {
  __shared__ float Wlds[64*129];
  __shared__ float lsum[128], lsq[128];
  __shared__ float sc[128], shv[128];
  const int tid = threadIdx.x;
  if (tid < 128) {
    lsum[tid] = 0.f; lsq[tid] = 0.f;
    if (do_transform) { sc[tid] = tsc[tid]; shv[tid] = tsh[tid]; }
  }
  const int wave = tid >> 5, lane = tid & 31;
  const int lrow = lane & 15;
  const int off2 = (lane >> 4) << 1;
  const int rowbase = blockIdx.x * 128 + wave * 16;
  const int myrow = rowbase + lrow;

  v8f acc[8];
#pragma unroll
  for (int t = 0; t < 8; ++t) acc[t] = v8f_zero();

  for (int kc = 0; kc < 2; ++kc) {
    const int kbase = kc * 64;
    __syncthreads();
    for (int i = tid; i < 64*128; i += 256) {
      const int r = i >> 7, c = i & 127;
      Wlds[r*129 + c] = W[(size_t)(kbase + r)*Hh + c];
    }
    __syncthreads();

    v2f afr[16];
    const float* arow = A + (size_t)myrow*Hh + kbase + off2;
#pragma unroll
    for (int ks = 0; ks < 16; ++ks) {
      float x0 = arow[ks*4], x1 = arow[ks*4 + 1];
      if (do_transform) {
        const int kc0 = kbase + ks*4 + off2;
        x0 = fmaxf(0.f, fmaf(x0, sc[kc0],   shv[kc0]));
        x1 = fmaxf(0.f, fmaf(x1, sc[kc0+1], shv[kc0+1]));
      }
      afr[ks].x = x0; afr[ks].y = x1;
    }
#pragma unroll
    for (int tn = 0; tn < 8; ++tn) {
      const int col = tn*16 + lrow;
#pragma unroll
      for (int ks = 0; ks < 16; ++ks) {
        v2f bfr;
        bfr.x = Wlds[(ks*4 + off2    )*129 + col];
        bfr.y = Wlds[(ks*4 + off2 + 1)*129 + col];
        acc[tn] = wmma4_f32(afr[ks], bfr, acc[tn]);
      }
    }
  }

  const int mofs = (lane >> 4) * 8;
#pragma unroll
  for (int tn = 0; tn < 8; ++tn) {
    const int col = tn*16 + lrow;
    const float bv = bias[col];
    float s = 0.f, q = 0.f;
#pragma unroll
    for (int i = 0; i < 8; ++i) {
      const float v = acc[tn][i] + bv;
      out[(size_t)(rowbase + mofs + i)*Hh + col] = v;
      s += v; q += v*v;
    }
    atomicAdd(&lsum[col], s);
    atomicAdd(&lsq[col], q);
  }
  __syncthreads();
  if (tid < 128) { atomicAdd(&gsum[tid], lsum[tid]); atomicAdd(&gsq[tid], lsq[tid]); }
}

// ---------------------------------------------------------------------------
// Fold BN stats into per-column scale/shift: x_hat = relu(x*scale+shift).
// stats layout per instance: [sum 128][sumsq 128][scale 128][shift 128]
__global__ void k_finalize(float* __restrict__ stats, const float* __restrict__ g,
                           const float* __restrict__ be, float invM) {
  const int c = threadIdx.x;                    // 128 threads
  const float m  = stats[c] * invM;
  const float var = stats[128 + c] * invM - m*m;
  const float rs = rsqrtf(var + 1e-5f);
  const float scv = g[c] * rs;
  stats[256 + c] = scv;
  stats[384 + c] = be[c] - scv * m;
}

__global__ void k_zero(float* __restrict__ p, int n) {
  const int i = blockIdx.x*256 + threadIdx.x;
  if (i < n) p[i] = 0.f;
}

// ---------------------------------------------------------------------------
// feature[t,b,n,c] = relu(bn2_1(z)), readout[t,b,c] = mean over n.
__global__ void k_output(const float* __restrict__ z, const float* __restrict__ tsc,
                         const float* __restrict__ tsh, float* __restrict__ feat,
                         float* __restrict__ readout) {
  __shared__ float red[256];
  const int bt = blockIdx.x;                    // bt = b*T + t
  const int b = bt / Tt, t = bt - b*Tt;
  const int tid = threadIdx.x;
  const int c = tid & 127;
  const float scv = tsc[c], shv = tsh[c];
  const float* zp = z + (size_t)bt*Nn*Hh;
  float* fp = feat + ((size_t)t*Bb + b)*Nn*Hh;
  float partial = 0.f;
  for (int i = tid; i < Nn*Hh; i += 256) {      // i = n*128 + c, c invariant
    const float v = fmaxf(0.f, fmaf(zp[i], scv, shv));
    fp[i] = v;
    partial += v;
  }
  red[tid] = partial;
  __syncthreads();
  if (tid < 128)
    readout[((size_t)t*Bb + b)*Hh + c] = (red[tid] + red[tid + 128]) * (1.f/(float)Nn);
}

// ---------------------------------------------------------------------------
extern "C" void kernel_launch(void* const* d_in, const int* in_sizes, int n_in,
                              void* d_out, int out_size, void* d_ws, size_t ws_size,
                              hipStream_t stream) {
  const float* v1  = (const float*)d_in[0];
  const float* a1  = (const float*)d_in[1];
  const float* W0  = (const float*)d_in[2];
  const float* b0  = (const float*)d_in[3];
  const float* eps = (const float*)d_in[4];
  const float* W1  = (const float*)d_in[5];
  const float* b1  = (const float*)d_in[6];
  const float* g1  = (const float*)d_in[7];
  const float* be1 = (const float*)d_in[8];
  const float* W2  = (const float*)d_in[9];
  const float* b2  = (const float*)d_in[10];
  const float* g2  = (const float*)d_in[11];
  const float* be2 = (const float*)d_in[12];

  float* feat = (float*)d_out;
  float* readout = feat + (size_t)Tt*Bb*Nn*Hh;

  const size_t nh = (size_t)MROWS * Hh;         // 26,214,400 floats per buffer
  float* bufA = (float*)d_ws;
  float* bufB = bufA + nh;
  unsigned* mask = (unsigned*)(bufB + nh);      // 2,662,400 u32 (10.6 MB)
  float* bmin = (float*)(mask + (size_t)BT*Nn*WPR);
  float* stats = bmin + BT;                     // 4 instances x 512 floats

  const float invM = 1.f / (float)MROWS;

  k_zero<<<8, 256, 0, stream>>>(stats, 4*512);
  k_blockmin<<<BT, 256, 0, stream>>>(a1, bmin);
  k_maskpack<<<BT, 256, 0, stream>>>(a1, bmin, mask);
  k_gemm0<<<MROWS/128, 256, 0, stream>>>(v1, W0, b0, bufA);

  // ---- layer 0 ----
  k_agg<<<dim3(BT, Hh/16), 256, 0, stream>>>(mask, bufA, bufB, eps, 0,
                                             nullptr, nullptr, 0);
  k_gemm_mlp<<<MROWS/128, 256, 0, stream>>>(bufB, W1, b1, bufA,
                                            nullptr, nullptr, 0,
                                            stats + 0, stats + 128);
  k_finalize<<<1, 128, 0, stream>>>(stats + 0, g1, be1, invM);
  k_gemm_mlp<<<MROWS/128, 256, 0, stream>>>(bufA, W2, b2, bufB,
                                            stats + 256, stats + 384, 1,
                                            stats + 512, stats + 640);
  k_finalize<<<1, 128, 0, stream>>>(stats + 512, g2, be2, invM);

  // ---- layer 1 ----
  k_agg<<<dim3(BT, Hh/16), 256, 0, stream>>>(mask, bufB, bufA, eps, 1,
                                             stats + 512 + 256, stats + 512 + 384, 1);
  k_gemm_mlp<<<MROWS/128, 256, 0, stream>>>(bufA, W1 + Hh*Hh, b1 + Hh, bufB,
                                            nullptr, nullptr, 0,
                                            stats + 1024, stats + 1152);
  k_finalize<<<1, 128, 0, stream>>>(stats + 1024, g1 + Hh, be1 + Hh, invM);
  k_gemm_mlp<<<MROWS/128, 256, 0, stream>>>(bufB, W2 + Hh*Hh, b2 + Hh, bufA,
                                            stats + 1024 + 256, stats + 1024 + 384, 1,
                                            stats + 1536, stats + 1664);
  k_finalize<<<1, 128, 0, stream>>>(stats + 1536, g2 + Hh, be2 + Hh, invM);

  k_output<<<BT, 256, 0, stream>>>(bufA, stats + 1536 + 256, stats + 1536 + 384,
                                   feat, readout);
}